// GraphEncoder_71691594105496
// MI455X (gfx1250) — compile-verified
//
#include <hip/hip_runtime.h>
#include <hip/hip_bf16.h>

#define N_NODES  50000
#define N_EDGES  640000
#define RANK     128
#define ORDER    3
#define N_GRAPHS 50
#define LN_EPS   1e-5f

typedef float v2f __attribute__((ext_vector_type(2)));
typedef float v8f __attribute__((ext_vector_type(8)));

// ---------------------------------------------------------------------------
// Utility: zero a buffer (float4 granularity)
// ---------------------------------------------------------------------------
__global__ void __launch_bounds__(256) zero4_kernel(float4* __restrict__ p, int n4) {
    int i = blockIdx.x * 256 + threadIdx.x;
    if (i < n4) p[i] = make_float4(0.f, 0.f, 0.f, 0.f);
}

// ---------------------------------------------------------------------------
// Degrees: atomic +1 per edge endpoint
// ---------------------------------------------------------------------------
__global__ void __launch_bounds__(256) degree_kernel(const int* __restrict__ src,
                                                     const int* __restrict__ dst,
                                                     float* __restrict__ deg_out,
                                                     float* __restrict__ deg_in,
                                                     int nedges) {
    int e = blockIdx.x * 256 + threadIdx.x;
    if (e < nedges) {
        atomicAdd(deg_out + src[e], 1.0f);
        atomicAdd(deg_in  + dst[e], 1.0f);
    }
}

// deg -> rsqrt(max(deg,1)) in place (applied over deg_out and deg_in contiguously)
__global__ void __launch_bounds__(256) invsqrt_kernel(float* __restrict__ p, int n) {
    int i = blockIdx.x * 256 + threadIdx.x;
    if (i < n) p[i] = rsqrtf(fmaxf(p[i], 1.0f));
}

// ---------------------------------------------------------------------------
// H = (X @ W) * inv_sqrt_out[row], via fp32 WMMA 16x16x4.
// Block: 256 threads = 8 waves; block covers 128 rows x 128 cols.
// Wave w covers rows [r0, r0+16), all 8 column tiles, K = 0..127 in steps of 4.
// ---------------------------------------------------------------------------
__global__ void __launch_bounds__(256) gemm_scale_kernel(const float* __restrict__ X,
                                                         const float* __restrict__ W,
                                                         const float* __restrict__ inv_out,
                                                         float* __restrict__ H,
                                                         int nrows) {
    // 128x128 fp32 W tile, XOR-swizzled per K-row pair to avoid bank conflicts
    __shared__ float ldsW[RANK * RANK]; // exactly 64 KB

    // Cooperative coalesced load of W, applying the swizzle: col ^= ((row&2)<<4)
    for (int i4 = threadIdx.x; i4 < RANK * RANK / 4; i4 += 256) {
        int row = i4 >> 5;            // /32 float4s per row
        int col = (i4 & 31) << 2;     // float4-aligned column
        float4 v = ((const float4*)W)[i4];
        int scol = col ^ ((row & 2) << 4);
        *(float4*)&ldsW[row * RANK + scol] = v;
    }
    __syncthreads();

    const int wave = threadIdx.x >> 5;
    const int lane = threadIdx.x & 31;
    const int half = lane >> 4;   // 0: lanes 0-15, 1: lanes 16-31
    const int n    = lane & 15;

    const int r0   = blockIdx.x * 128 + wave * 16;
    const int mrow = r0 + n;                       // A-row this lane loads
    const int arow = (mrow < nrows) ? mrow : 0;    // clamp: EXEC must stay all-ones
    const float* Xrow = X + (size_t)arow * RANK;

    v8f acc[8] = {};

    #pragma unroll 4
    for (int ks = 0; ks < RANK / 4; ++ks) {
        const int k = ks * 4 + 2 * half;           // this lane's K base (even)
        // A fragment: 16x4, lane<16 holds K=k..k+1 of row m; lane>=16 holds K=k+2..k+3
        const float2 af = *(const float2*)(Xrow + k);
        v2f a; a.x = af.x; a.y = af.y;

        const int swz = (k & 2) << 4;              // LDS bank swizzle for this K row-pair
        #pragma unroll
        for (int t = 0; t < 8; ++t) {
            v2f b;
            b.x = ldsW[(k    ) * RANK + ((t * 16 + n) ^ swz)];
            b.y = ldsW[(k + 1) * RANK + ((t * 16 + n) ^ swz)];
            acc[t] = __builtin_amdgcn_wmma_f32_16x16x4_f32(
                false, a, false, b, (short)0, acc[t], false, false);
        }
    }

    // C/D layout: VGPR v holds row (v + 8*half), column n, per 16-wide tile t
    #pragma unroll
    for (int v = 0; v < 8; ++v) {
        const int row = r0 + v + 8 * half;
        if (row < nrows) {
            const float s = inv_out[row];
            float* Hrow = H + (size_t)row * RANK;
            #pragma unroll
            for (int t = 0; t < 8; ++t) {
                Hrow[t * 16 + n] = acc[t][v] * s;
            }
        }
    }
}

// ---------------------------------------------------------------------------
// agg[dst[e]] += h[src[e]] : one wave per edge, float4 per lane (128 floats/row)
// ---------------------------------------------------------------------------
__global__ void __launch_bounds__(256) scatter_kernel(const float* __restrict__ H,
                                                      const int* __restrict__ src,
                                                      const int* __restrict__ dst,
                                                      float* __restrict__ agg,
                                                      int nedges) {
    int gid  = blockIdx.x * 256 + threadIdx.x;
    int edge = gid >> 5;
    int lane = threadIdx.x & 31;
    if (edge < nedges) {
        int s = src[edge];
        int d = dst[edge];
        const float4 v = *(const float4*)(H + (size_t)s * RANK + lane * 4);
        float* p = agg + (size_t)d * RANK + lane * 4;
        atomicAdd(p + 0, v.x);
        atomicAdd(p + 1, v.y);
        atomicAdd(p + 2, v.z);
        atomicAdd(p + 3, v.w);
    }
}

// ---------------------------------------------------------------------------
// out = relu(LayerNorm(agg * inv_sqrt_in + b) * gamma + beta)
// One wave per node; wave32 shuffle reduction for mean / variance.
// ---------------------------------------------------------------------------
__global__ void __launch_bounds__(256) post_kernel(const float* __restrict__ agg,
                                                   const float* __restrict__ inv_in,
                                                   const float* __restrict__ bias,
                                                   const float* __restrict__ gamma,
                                                   const float* __restrict__ beta,
                                                   float* __restrict__ out,
                                                   int nrows) {
    int node = (blockIdx.x * 256 + threadIdx.x) >> 5;
    int lane = threadIdx.x & 31;
    if (node >= nrows) return;

    const float s = inv_in[node];
    const int c0  = lane * 4;
    const float4 v = *(const float4*)(agg + (size_t)node * RANK + c0);

    float x0 = v.x * s + bias[c0 + 0];
    float x1 = v.y * s + bias[c0 + 1];
    float x2 = v.z * s + bias[c0 + 2];
    float x3 = v.w * s + bias[c0 + 3];

    float sum = x0 + x1 + x2 + x3;
    float sq  = x0 * x0 + x1 * x1 + x2 * x2 + x3 * x3;
    #pragma unroll
    for (int off = 16; off > 0; off >>= 1) {
        sum += __shfl_xor(sum, off, 32);
        sq  += __shfl_xor(sq,  off, 32);
    }
    const float mu   = sum * (1.0f / RANK);
    const float var  = sq * (1.0f / RANK) - mu * mu;
    const float rstd = rsqrtf(var + LN_EPS);

    float4 o;
    o.x = fmaxf((x0 - mu) * rstd * gamma[c0 + 0] + beta[c0 + 0], 0.f);
    o.y = fmaxf((x1 - mu) * rstd * gamma[c0 + 1] + beta[c0 + 1], 0.f);
    o.z = fmaxf((x2 - mu) * rstd * gamma[c0 + 2] + beta[c0 + 2], 0.f);
    o.w = fmaxf((x3 - mu) * rstd * gamma[c0 + 3] + beta[c0 + 3], 0.f);
    *(float4*)(out + (size_t)node * RANK + c0) = o;
}

// ---------------------------------------------------------------------------
// first_idx = exclusive cumsum(batch_num_nodes); out[g] = feats[first_idx[g]]
// ---------------------------------------------------------------------------
__global__ void __launch_bounds__(256) gather_kernel(const float* __restrict__ feats,
                                                     const int* __restrict__ bnn,
                                                     float* __restrict__ out) {
    __shared__ int first[N_GRAPHS];
    if (threadIdx.x == 0) {
        int acc = 0;
        for (int g = 0; g < N_GRAPHS; ++g) { first[g] = acc; acc += bnn[g]; }
    }
    __syncthreads();
    for (int t = threadIdx.x; t < N_GRAPHS * RANK; t += 256) {
        int g = t >> 7;
        int c = t & (RANK - 1);
        out[t] = feats[(size_t)first[g] * RANK + c];
    }
}

// ---------------------------------------------------------------------------
// Host launcher
// ---------------------------------------------------------------------------
extern "C" void kernel_launch(void* const* d_in, const int* in_sizes, int n_in,
                              void* d_out, int out_size, void* d_ws, size_t ws_size,
                              hipStream_t stream) {
    (void)in_sizes; (void)n_in; (void)out_size; (void)ws_size;

    const float* features = (const float*)d_in[0];
    const int*   src      = (const int*)d_in[1];
    const int*   dst      = (const int*)d_in[2];
    const int*   bnn      = (const int*)d_in[3];
    const float* Ws       = (const float*)d_in[4];
    const float* bs       = (const float*)d_in[5];
    const float* gammas   = (const float*)d_in[6];
    const float* betas    = (const float*)d_in[7];
    float*       out      = (float*)d_out;

    const size_t NR = (size_t)N_NODES * RANK;
    float* feats_buf = (float*)d_ws;        // 25.6 MB
    float* h_buf     = feats_buf + NR;      // 25.6 MB
    float* agg_buf   = h_buf + NR;          // 25.6 MB
    float* deg       = agg_buf + NR;        // deg_out[N] then deg_in[N] (contiguous)
    float* deg_out   = deg;
    float* deg_in    = deg + N_NODES;

    // Degrees -> inverse sqrt factors (recomputed every call: deterministic)
    zero4_kernel<<<(2 * N_NODES / 4 + 255) / 256, 256, 0, stream>>>((float4*)deg, 2 * N_NODES / 4);
    degree_kernel<<<(N_EDGES + 255) / 256, 256, 0, stream>>>(src, dst, deg_out, deg_in, N_EDGES);
    invsqrt_kernel<<<(2 * N_NODES + 255) / 256, 256, 0, stream>>>(deg, 2 * N_NODES);

    const int gemm_grid    = (N_NODES + 127) / 128;                 // 391
    const int zero_grid    = (int)((NR / 4 + 255) / 256);           // 6250
    const int scatter_grid = (int)(((size_t)N_EDGES * 32 + 255) / 256);  // 80000
    const int node_grid    = (int)(((size_t)N_NODES * 32 + 255) / 256);  // 6250

    const float* cur = features;
    for (int layer = 0; layer < ORDER; ++layer) {
        gemm_scale_kernel<<<gemm_grid, 256, 0, stream>>>(
            cur, Ws + (size_t)layer * RANK * RANK, deg_out, h_buf, N_NODES);
        zero4_kernel<<<zero_grid, 256, 0, stream>>>((float4*)agg_buf, (int)(NR / 4));
        scatter_kernel<<<scatter_grid, 256, 0, stream>>>(h_buf, src, dst, agg_buf, N_EDGES);
        post_kernel<<<node_grid, 256, 0, stream>>>(
            agg_buf, deg_in, bs + layer * RANK, gammas + layer * RANK,
            betas + layer * RANK, feats_buf, N_NODES);
        cur = feats_buf;
    }

    gather_kernel<<<1, 256, 0, stream>>>(feats_buf, bnn, out);
}